// Unresample_45561013076304
// MI455X (gfx1250) — compile-verified
//
#include <hip/hip_runtime.h>
#include <stdint.h>

// Problem constants (from reference)
#define H_IN   512
#define W_IN   1024
#define NPLANES 128                    // B*C = 4*32
#define PLANE_ELEMS (H_IN * W_IN)      // 524288 (== H_OUT*W_OUT here)
#define OUT_PIX (512 * 1024)           // H_OUT * W_OUT
#define BLOCK  256
#define PPT    4                       // pixels per thread
#define PIX_PER_BLOCK (BLOCK * PPT)    // 1024 pixels -> 8KB of sample_map

typedef __attribute__((ext_vector_type(4))) float         f4;
typedef __attribute__((ext_vector_type(4))) unsigned int  u32x4;
typedef __attribute__((ext_vector_type(8))) int           i32x8;
typedef __attribute__((ext_vector_type(4))) int           i32x4;

__global__ __launch_bounds__(BLOCK) void resample_kernel(
    const float* __restrict__ x,      // [128][512*1024]
    const float* __restrict__ smap,   // [512*1024][2]
    float* __restrict__ out)          // [128][512*1024]
{
  __shared__ float scoord[PIX_PER_BLOCK * 2];   // 8 KB staged via TDM

  const int t = threadIdx.x;
  const int blockPixBase = blockIdx.x * PIX_PER_BLOCK;

  // ---- TDM: async-copy this block's contiguous sample_map slice into LDS ----
  // One wave issues the descriptor (TDM ignores EXEC; avoid 8x redundant DMAs).
  if (t < 32) {
    const uint64_t gaddr = (uint64_t)(uintptr_t)(smap + (size_t)blockPixBase * 2);
    const uint32_t lds_addr = (uint32_t)(uintptr_t)(&scoord[0]); // low 32 bits of LDS aperture = LDS byte offset
    const uint32_t n_dw = (uint32_t)(PIX_PER_BLOCK * 2);         // 2048 dwords

    // D# group 0 (128b): count=1 | lds_addr | global_addr[56:0] | type=2
    u32x4 g0;
    g0.x = 1u;
    g0.y = lds_addr;
    g0.z = (uint32_t)(gaddr & 0xFFFFFFFFu);
    g0.w = (uint32_t)((gaddr >> 32) & 0x1FFFFFFu) | (2u << 30);

    // D# group 1 (256b): data_size=4B; tensor 2048x1; tile 2048x1; stride0=2048
    i32x8 g1;
    g1[0] = (int)(2u << 16);                                   // wg_mask=0, data_size=2(4B)
    g1[1] = (int)((n_dw & 0xFFFFu) << 16);                     // tensor_dim0[15:0]
    g1[2] = (int)(((n_dw >> 16) & 0xFFFFu) | (1u << 16));      // tensor_dim0[31:16], tensor_dim1=1
    g1[3] = (int)(n_dw << 16);                                 // tile_dim0 = 2048
    g1[4] = 1;                                                 // tile_dim1 = 1, tile_dim2 = 0
    g1[5] = (int)n_dw;                                         // tensor_dim0_stride lo32
    g1[6] = (int)((n_dw & 0xFFFFu) << 16);                     // stride0 hi=0, stride1 lo16
    g1[7] = 0;

    i32x4 z4 = {0, 0, 0, 0};
#if __clang_major__ >= 23
    i32x8 z8 = {0, 0, 0, 0, 0, 0, 0, 0};
    __builtin_amdgcn_tensor_load_to_lds(g0, g1, z4, z4, z8, 0);
#else
    __builtin_amdgcn_tensor_load_to_lds(g0, g1, z4, z4, 0);
#endif
    __builtin_amdgcn_s_wait_tensorcnt(0);
  }
  __syncthreads();

  // ---- Per-pixel setup (done once, reused across all 128 planes) ----
  int   i00[PPT], i10[PPT], i01[PPT], i11[PPT];
  float w00[PPT], w10[PPT], w01[PPT], w11[PPT];
#pragma unroll
  for (int j = 0; j < PPT; ++j) {
    const float sx = scoord[(t * PPT + j) * 2 + 0];
    const float sy = scoord[(t * PPT + j) * 2 + 1];
    const float fx0 = floorf(sx), fy0 = floorf(sy);
    const int x0 = (int)fx0, y0 = (int)fy0;
    const float dx = sx - fx0, dy = sy - fy0;
    const float ex = 1.0f - dx, ey = 1.0f - dy;

    const bool vx0 = (x0 >= 0) && (x0 < W_IN);
    const bool vy0 = (y0 >= 0) && (y0 < H_IN);
    const bool vx1 = (x0 + 1 >= 0) && (x0 + 1 < W_IN);
    const bool vy1 = (y0 + 1 >= 0) && (y0 + 1 < H_IN);

    const int cx0 = min(max(x0, 0), W_IN - 1);
    const int cy0 = min(max(y0, 0), H_IN - 1);
    const int cx1 = min(max(x0 + 1, 0), W_IN - 1);
    const int cy1 = min(max(y0 + 1, 0), H_IN - 1);

    i00[j] = cy0 * W_IN + cx0;
    i10[j] = cy0 * W_IN + cx1;
    i01[j] = cy1 * W_IN + cx0;
    i11[j] = cy1 * W_IN + cx1;
    w00[j] = ex * ey * (float)(vx0 && vy0);
    w10[j] = dx * ey * (float)(vx1 && vy0);
    w01[j] = ex * dy * (float)(vx0 && vy1);
    w11[j] = dx * dy * (float)(vx1 && vy1);
  }

  // ---- Sweep all 128 planes; gathers hit L2 (planes swept in lockstep),
  //      output streamed with non-temporal 128-bit stores ----
  const size_t pixOff = (size_t)blockPixBase + (size_t)t * PPT;
#pragma unroll 4
  for (int p = 0; p < NPLANES; ++p) {
    const float* __restrict__ pl = x + (size_t)p * PLANE_ELEMS;
    f4 r;
#pragma unroll
    for (int j = 0; j < PPT; ++j) {
      r[j] = pl[i00[j]] * w00[j] + pl[i10[j]] * w10[j] +
             pl[i01[j]] * w01[j] + pl[i11[j]] * w11[j];
    }
    __builtin_nontemporal_store(r, (f4*)(out + (size_t)p * PLANE_ELEMS + pixOff));
  }
}

extern "C" void kernel_launch(void* const* d_in, const int* in_sizes, int n_in,
                              void* d_out, int out_size, void* d_ws, size_t ws_size,
                              hipStream_t stream) {
  (void)in_sizes; (void)n_in; (void)out_size; (void)d_ws; (void)ws_size;
  const float* x    = (const float*)d_in[0];
  const float* smap = (const float*)d_in[1];
  float* out        = (float*)d_out;
  const int blocks = OUT_PIX / PIX_PER_BLOCK;  // 512
  resample_kernel<<<dim3(blocks), dim3(BLOCK), 0, stream>>>(x, smap, out);
}